// OptNetEq_37082747633748
// MI455X (gfx1250) — compile-verified
//
#include <hip/hip_runtime.h>

typedef __attribute__((ext_vector_type(2))) float v2f;
typedef __attribute__((ext_vector_type(8))) float v8f;

// -------- prep: build K = [[I6, A'],[A, 0]], invert (Gauss-Jordan, f64),
// store W = Kinv[:6,:6] (ws[0..35], row-major) and c[j] = -puzzles[3] *
// (Kinv[j][6]+Kinv[j][7]+Kinv[j][8]) (ws[36..41]). One thread; tiny. --------
__global__ void optnet_prep_kernel(const float* __restrict__ puzzles,
                                   const float* __restrict__ A,
                                   float* __restrict__ ws) {
  if (threadIdx.x != 0 || blockIdx.x != 0) return;
  double M[9][18];
  for (int i = 0; i < 9; ++i)
    for (int j = 0; j < 18; ++j) M[i][j] = 0.0;
  for (int i = 0; i < 6; ++i) M[i][i] = 1.0;               // Q = I6
  for (int r = 0; r < 3; ++r)
    for (int j = 0; j < 6; ++j) {
      double a = (double)A[r * 6 + j];
      M[j][6 + r] = a;                                      // A^T block
      M[6 + r][j] = a;                                      // A block
    }
  for (int i = 0; i < 9; ++i) M[i][9 + i] = 1.0;            // augment I9

  for (int col = 0; col < 9; ++col) {
    int piv = col;
    double best = M[col][col] < 0.0 ? -M[col][col] : M[col][col];
    for (int r = col + 1; r < 9; ++r) {
      double v = M[r][col] < 0.0 ? -M[r][col] : M[r][col];
      if (v > best) { best = v; piv = r; }
    }
    if (piv != col)
      for (int j = 0; j < 18; ++j) {
        double t = M[col][j]; M[col][j] = M[piv][j]; M[piv][j] = t;
      }
    double inv = 1.0 / M[col][col];
    for (int j = 0; j < 18; ++j) M[col][j] *= inv;
    for (int r = 0; r < 9; ++r) {
      if (r == col) continue;
      double f = M[r][col];
      if (f != 0.0)
        for (int j = 0; j < 18; ++j) M[r][j] -= f * M[col][j];
    }
  }

  double bval = -(double)puzzles[3];                        // b = p[0,3] = -puzzles[0,3]
  for (int j = 0; j < 6; ++j) {
    for (int k = 0; k < 6; ++k) ws[j * 6 + k] = (float)M[j][9 + k];
    ws[36 + j] = (float)(bval * (M[j][9 + 6] + M[j][9 + 7] + M[j][9 + 8]));
  }
}

// -------- main: y[16-row tile] = Q(16x6) * W^T(6x6 pad->8x16) + c, via two
// V_WMMA_F32_16X16X4_F32 (K = 0..3 then 4..5 + zero pad). 4 tiles per wave.
// A layout (ISA 16x4 f32): lane<16 -> M=lane, K=0/1; lane>=16 -> M=lane-16, K=2/3.
// B layout (4x16 f32, mirrored): VGPR0 = K={0|2}, VGPR1 = K={1|3} per lane half.
// D layout: VGPR r -> row (hi*8 + r), col = lane&15. --------
__global__ void __launch_bounds__(256) optnet_main_kernel(
    const float* __restrict__ in, float* __restrict__ out,
    const float* __restrict__ ws, int numTiles) {
  const int lane = threadIdx.x & 31;
  const int waveId = (blockIdx.x << 3) | (threadIdx.x >> 5);
  const int n  = lane & 15;        // output column / row-in-tile for A
  const int hi = lane >> 4;        // lane half selects K pair

  // B operand + per-column constant (wave-invariant across tiles).
  const int nn = (n < 6) ? n : 0;  // clamp reads; zero below
  const float* wrow = ws + nn * 6; // W[nn][*] ; B[K][n] = W^T[K][n] = W[n][K]
  float b00 = wrow[hi ? 2 : 0];
  float b01 = wrow[hi ? 3 : 1];
  float b10 = hi ? 0.0f : wrow[4]; // K=6,7 pad -> 0
  float b11 = hi ? 0.0f : wrow[5];
  float cn  = ws[36 + nn];
  if (n >= 6) { b00 = 0.0f; b01 = 0.0f; b10 = 0.0f; b11 = 0.0f; cn = 0.0f; }
  v2f bLo = { b00, b01 };
  v2f bHi = { b10, b11 };
  const v2f vzero = { 0.0f, 0.0f };

  const int tile0 = waveId * 4;
  #pragma unroll
  for (int t = 0; t < 4; ++t) {
    int tile = tile0 + t;
    if (tile >= numTiles) break;             // wave-uniform: EXEC stays full
    int base = tile << 4;                    // first batch row of tile
    int row  = base + n;                     // this lane's A-matrix row (M = n)
    const float* rowp = in + (size_t)row * 6;

    v2f a0 = *(const v2f*)(rowp + (hi << 1));        // K = 2*hi, 2*hi+1
    if (hi && row < 2) a0.y = 0.0f;                  // zero q[row][3] for rows 0,1
    v2f a1 = hi ? vzero : *(const v2f*)(rowp + 4);   // K = 4,5 (6,7 padded)

    v8f acc = {};
    acc = __builtin_amdgcn_wmma_f32_16x16x4_f32(false, a0, false, bLo,
                                                (short)0, acc, false, false);
    acc = __builtin_amdgcn_wmma_f32_16x16x4_f32(false, a1, false, bHi,
                                                (short)0, acc, false, false);

    if (n < 6) {                             // lanes n>=6 hold padding
      float* orow = out + (size_t)(base + (hi << 3)) * 6 + n;
      #pragma unroll
      for (int r = 0; r < 8; ++r) orow[(size_t)r * 6] = acc[r] + cn;
    }
  }
}

extern "C" void kernel_launch(void* const* d_in, const int* in_sizes, int n_in,
                              void* d_out, int out_size, void* d_ws, size_t ws_size,
                              hipStream_t stream) {
  const float* puzzles = (const float*)d_in[0];
  const float* A       = (const float*)d_in[1];
  float* out = (float*)d_out;
  float* ws  = (float*)d_ws;

  int nRows    = in_sizes[0] / 6;            // B = 4,194,304
  int numTiles = (nRows + 15) / 16;          // 16 rows per WMMA tile

  optnet_prep_kernel<<<1, 32, 0, stream>>>(puzzles, A, ws);

  int waves  = (numTiles + 3) / 4;           // 4 tiles per wave
  int blocks = (waves + 7) / 8;              // 8 waves per 256-thread block
  optnet_main_kernel<<<blocks, 256, 0, stream>>>(puzzles, out, ws, numTiles);
}